// GRU_17446157156705
// MI455X (gfx1250) — compile-verified
//
#include <hip/hip_runtime.h>
#include <hip/hip_bf16.h>
#include <math.h>

// GRU scan, D=2048, T=1024, fp32 throughout (matches reference precision).
// Kernel 1: input projections via native fp32 WMMA (v_wmma_f32_16x16x4_f32),
//           32x64 register block per wave (2 M-tiles x 4 N-tiles).
// Kernel 2: persistent sequential scan; recurrent weights streamed from L2
//           (all 6 weight matrices = 100 MB < 192 MB L2), software prefetch
//           (global_prefetch_b8) covering the 8 KB-stride weight walk.

#define DMODEL 2048
#define TSEQ   1024

#define NBLK        64                     // persistent blocks in scan kernel
#define SCAN_THR    256                    // 8 wave32s per block
#define DPB         (DMODEL / NBLK)        // 32 output channels per block
#define KPW         (DMODEL / (SCAN_THR/32)) // 256 K-elements per wave
#define PFDIST      16                     // prefetch distance (k-rows)

#define MT 2                               // M tiles per wave (32 rows)
#define NT 4                               // N tiles per wave (64 cols)

typedef float v2f __attribute__((ext_vector_type(2)));
typedef float v8f __attribute__((ext_vector_type(8)));

// ---------------------------------------------------------------------------
// Init: zero grid-barrier state, seed h ping-pong buffer with last_state.
// Re-run on every kernel_launch call -> no reliance on stale ws state.
// ---------------------------------------------------------------------------
__global__ void gru_init(const float* __restrict__ h0,
                         float* __restrict__ hbuf,       // 2*DMODEL floats
                         unsigned* __restrict__ bar) {   // bar[0]=count bar[1]=gen
  int i = blockIdx.x * blockDim.x + threadIdx.x;
  if (i < DMODEL) hbuf[i] = h0[i];
  if (i == 0) { bar[0] = 0u; bar[1] = 0u; }
}

// ---------------------------------------------------------------------------
// Input projections: O_g = X @ W_g + b_g for g in {r,z,n}.
// Each wave owns a 32x64 C block = MT x NT 16x16 tiles; K-loop of
// v_wmma_f32_16x16x4_f32 with A-frag reuse across NT and B-frag reuse
// across MT.
//
// fp32 16x4 A layout (ISA 7.12.2): lanes 0-15 hold M=lane, v[0]=K0,v[1]=K1;
// lanes 16-31 hold M=lane-16, v[0]=K2,v[1]=K3.  B (4x16) mirrored with
// lane = N and lane-half selecting the K pair.  C/D: VGPR j holds row
// M=j (lanes 0-15) / M=j+8 (lanes 16-31), N = lane&15.
// ---------------------------------------------------------------------------
__global__ __launch_bounds__(256) void gru_input_proj(
    const float* __restrict__ X,
    const float* __restrict__ Wr, const float* __restrict__ Wz,
    const float* __restrict__ Wn,
    const float* __restrict__ br, const float* __restrict__ bz,
    const float* __restrict__ bn,
    float* __restrict__ Or, float* __restrict__ Oz, float* __restrict__ On) {
  const int wave = blockIdx.x * (blockDim.x >> 5) + (threadIdx.x >> 5);
  const int lane = threadIdx.x & 31;
  const int nGrpN = DMODEL / (16 * NT);          // 32
  const int nGrpM = TSEQ / (16 * MT);            // 32
  const int perMat = nGrpM * nGrpN;              // 1024 wave-jobs per matrix
  const int mat = wave / perMat;
  if (mat >= 3) return;
  const int rem = wave - mat * perMat;
  const int t0 = (rem / nGrpN) * (16 * MT);
  const int n0 = (rem % nGrpN) * (16 * NT);

  const float* __restrict__ W = (mat == 0) ? Wr : (mat == 1) ? Wz : Wn;
  const float* __restrict__ b = (mat == 0) ? br : (mat == 1) ? bz : bn;
  float* __restrict__ O       = (mat == 0) ? Or : (mat == 1) ? Oz : On;

  const int l15 = lane & 15;
  const int hi  = lane >> 4;        // 0 for lanes 0-15, 1 for lanes 16-31

  v8f acc[MT][NT];
#pragma unroll
  for (int m = 0; m < MT; ++m)
#pragma unroll
    for (int j = 0; j < NT; ++j) acc[m][j] = (v8f){};

  for (int k = 0; k < DMODEL; k += 4) {
    const int kk = k + (hi << 1);   // K pair selected by lane half
    v2f a[MT];
#pragma unroll
    for (int m = 0; m < MT; ++m)
      a[m] = *(const v2f*)(X + (size_t)(t0 + m * 16 + l15) * DMODEL + kk);
#pragma unroll
    for (int j = 0; j < NT; ++j) {
      const int bcol = n0 + j * 16 + l15;
      v2f bf;
      bf.x = W[(size_t)kk       * DMODEL + bcol];
      bf.y = W[(size_t)(kk + 1) * DMODEL + bcol];
#pragma unroll
      for (int m = 0; m < MT; ++m)
        acc[m][j] = __builtin_amdgcn_wmma_f32_16x16x4_f32(
            false, a[m], false, bf, (short)0, acc[m][j], false, false);
    }
  }

#pragma unroll
  for (int j = 0; j < NT; ++j) {
    const int col = n0 + j * 16 + l15;
    const float bias = b[col];
#pragma unroll
    for (int m = 0; m < MT; ++m) {
      const int rowbase = t0 + m * 16 + (hi << 3);
#pragma unroll
      for (int r = 0; r < 8; ++r)
        O[(size_t)(rowbase + r) * DMODEL + col] = acc[m][j][r] + bias;
    }
  }
}

// ---------------------------------------------------------------------------
// Sequential scan. 64 persistent blocks, grid barrier per timestep.
// Block b owns output channels [b*32, b*32+32) for all three gates.
// h_eff (reset-masked) staged in LDS; each wave covers a 256-wide K slice
// with fully coalesced weight reads (L2-resident working set) plus a
// software prefetch to hide the 8 KB row stride.
// ---------------------------------------------------------------------------
__global__ __launch_bounds__(SCAN_THR) void gru_scan(
    const float* __restrict__ Xr, const float* __restrict__ Xz,
    const float* __restrict__ Xn,
    const float* __restrict__ Whr, const float* __restrict__ Whz,
    const float* __restrict__ Whn,
    const float* __restrict__ bhn,
    const int*   __restrict__ term,
    float* __restrict__ hbuf,          // 2*DMODEL ping-pong
    unsigned* __restrict__ bar,        // [0]=arrive count, [1]=generation
    float* __restrict__ y) {           // 2*T*D (y_t then new_states)
  __shared__ float hs[DMODEL];         // 8 KB: masked h_{t-1}
  __shared__ float red[SCAN_THR];      // cross-wave reduction
  __shared__ float gdot[3][DPB];       // per-gate recurrent dots

  const int tid   = threadIdx.x;
  const int kgrp  = tid >> 5;          // wave id 0..7 -> K slice
  const int dlane = tid & 31;          // output channel within block
  const int d0    = blockIdx.x * DPB;

  for (int t = 0; t < TSEQ; ++t) {
    const float* __restrict__ hprev = hbuf + (size_t)(t & 1) * DMODEL;
    float* __restrict__ hnext       = hbuf + (size_t)((t + 1) & 1) * DMODEL;
    const int tm = term[t];

    for (int i = tid; i < DMODEL; i += SCAN_THR)
      hs[i] = tm ? 0.0f : hprev[i];
    __syncthreads();

    const float* __restrict__ Ws[3] = {Whr, Whz, Whn};
#pragma unroll
    for (int g = 0; g < 3; ++g) {
      const float* __restrict__ W = Ws[g];
      const size_t base = (size_t)(kgrp * KPW) * DMODEL + d0 + dlane;
      float acc = 0.0f;
#pragma unroll 4
      for (int k = 0; k < KPW; ++k) {
        if (k + PFDIST < KPW)          // global_prefetch_b8 over 8KB stride
          __builtin_prefetch(&W[base + (size_t)(k + PFDIST) * DMODEL], 0, 1);
        acc += hs[kgrp * KPW + k] * W[base + (size_t)k * DMODEL];
      }
      red[tid] = acc;
      __syncthreads();
      if (tid < DPB) {
        float s = 0.0f;
#pragma unroll
        for (int w = 0; w < SCAN_THR / 32; ++w) s += red[w * 32 + tid];
        gdot[g][tid] = s;
      }
      __syncthreads();
    }

    if (tid < DPB) {
      const int d = d0 + tid;
      const size_t off = (size_t)t * DMODEL + d;
      const float hv = hs[d];
      const float r  = 1.0f / (1.0f + expf(-(Xr[off] + gdot[0][tid])));
      const float z  = 1.0f / (1.0f + expf(-(Xz[off] + gdot[1][tid])));
      const float n  = tanhf(Xn[off] + r * (gdot[2][tid] + bhn[d]));
      const float nh = (1.0f - z) * n + z * hv;
      hnext[d] = nh;
      y[off] = nh;                                       // y_t
      y[(size_t)TSEQ * DMODEL + off] = nh;               // new_states
    }

    // -------- grid barrier (two-counter, generation-based) --------
    __threadfence();
    __syncthreads();
    if (tid == 0) {
      const unsigned target = (unsigned)(t + 1);
      const unsigned old = atomicAdd(&bar[0], 1u);
      if (old == (unsigned)(gridDim.x - 1)) {
        atomicExch(&bar[0], 0u);
        __threadfence();
        atomicAdd(&bar[1], 1u);
      } else {
        while (__atomic_load_n((const unsigned*)&bar[1], __ATOMIC_RELAXED) <
               target) {
          __builtin_amdgcn_s_sleep(2);
        }
      }
    }
    __syncthreads();
  }
}

// ---------------------------------------------------------------------------
extern "C" void kernel_launch(void* const* d_in, const int* in_sizes, int n_in,
                              void* d_out, int out_size, void* d_ws,
                              size_t ws_size, hipStream_t stream) {
  const float* X    = (const float*)d_in[0];   // [T,D]
  const int*   term = (const int*)  d_in[1];   // [T]
  const float* h0   = (const float*)d_in[2];   // [D]
  const float* Wir  = (const float*)d_in[3];
  const float* Wiz  = (const float*)d_in[4];
  const float* Win  = (const float*)d_in[5];
  const float* bir  = (const float*)d_in[6];
  const float* biz  = (const float*)d_in[7];
  const float* bin  = (const float*)d_in[8];
  const float* Whr  = (const float*)d_in[9];
  const float* Whz  = (const float*)d_in[10];
  const float* Whn  = (const float*)d_in[11];
  const float* bhn  = (const float*)d_in[12];

  float* ws = (float*)d_ws;
  float* Xr = ws;                                    // T*D
  float* Xz = Xr + (size_t)TSEQ * DMODEL;            // T*D
  float* Xn = Xz + (size_t)TSEQ * DMODEL;            // T*D
  float* hbuf = Xn + (size_t)TSEQ * DMODEL;          // 2*D
  unsigned* bar = (unsigned*)(hbuf + 2 * DMODEL);    // 2 words

  gru_init<<<(DMODEL + 255) / 256, 256, 0, stream>>>(h0, hbuf, bar);

  // 3 matrices x 32 M-groups x 32 N-groups = 3072 wave-jobs, 8 waves/block
  const int waveJobs = 3 * (TSEQ / (16 * MT)) * (DMODEL / (16 * NT));
  gru_input_proj<<<waveJobs / 8, 256, 0, stream>>>(X, Wir, Wiz, Win, bir, biz,
                                                   bin, Xr, Xz, Xn);

  gru_scan<<<NBLK, SCAN_THR, 0, stream>>>(Xr, Xz, Xn, Whr, Whz, Whn, bhn,
                                          term, hbuf, bar, (float*)d_out);
}